// NeiborAssigner1_6296422056680
// MI455X (gfx1250) — compile-verified
//
#include <hip/hip_runtime.h>
#include <math.h>
#include <stdint.h>

#define NFEAT 256
#define NHID  64

typedef float v2f __attribute__((ext_vector_type(2)));
typedef float v8f __attribute__((ext_vector_type(8)));

// ---------------- Stage 1a: deg[i] = 1 (self loop) ----------------
__global__ void deg_init_kernel(float* __restrict__ deg, int n) {
  int i = blockIdx.x * blockDim.x + threadIdx.x;
  if (i < n) deg[i] = 1.0f;
}

// ---------------- Stage 1b: deg[col[e]] += 1 ----------------------
__global__ void deg_count_kernel(const int* __restrict__ col,
                                 float* __restrict__ deg, int e) {
  int i = blockIdx.x * blockDim.x + threadIdx.x;
  if (i < e) atomicAdd(&deg[col[i]], 1.0f);
}

// ---------------- Stage 2: h = x @ W1 via V_WMMA_F32_16X16X4_F32 --
// Block = 256 threads = 8 waves; block computes 128 rows x 64 cols of h.
// W1 is staged in LDS as k-pairs: sW2[kp*64+n] = {W1[2kp][n], W1[2kp+1][n]}.
// Tail rows: A-row m only feeds D-row m, and tail D-rows are not stored,
// so A loads use a clamped row index and run UNCONDITIONALLY (no EXEC dance).
__global__ __launch_bounds__(256)
void gemm_x_w1_kernel(const float* __restrict__ x,
                      const float* __restrict__ W1,
                      float* __restrict__ h, int n) {
  __shared__ float sW[128 * NHID];          // 32 KB chunk (128 K-rows x 64)
  const int tid  = threadIdx.x;
  const int wave = tid >> 5;
  const int lane = tid & 31;
  const int l15  = lane & 15;
  const int kh   = lane >> 4;               // which K-pair this half-wave owns
  const int rowBase = blockIdx.x * 128 + wave * 16;
  const int row  = rowBase + l15;
  const int rowC = (row < n) ? row : (n - 1);   // clamped, always valid

  v8f acc[4];
#pragma unroll
  for (int t = 0; t < 4; ++t) acc[t] = (v8f){0,0,0,0,0,0,0,0};

  const float* xrow = x + (size_t)rowC * NFEAT + 2 * kh;
  __builtin_prefetch(xrow + 128, 0, 3);     // pull second K-chunk toward L0/L2

  for (int kc = 0; kc < 2; ++kc) {          // two 128-K chunks of W1
    __syncthreads();
    for (int idx = tid; idx < 64 * NHID; idx += 256) {   // 4096 float2 slots
      int kp = idx >> 6;                    // pair index 0..63
      int nn = idx & 63;
      int kg = kc * 128 + kp * 2;
      ((float2*)sW)[kp * NHID + nn] =
          make_float2(W1[(size_t)kg * NHID + nn], W1[(size_t)(kg + 1) * NHID + nn]);
    }
    __syncthreads();

#pragma unroll 4
    for (int k0 = 0; k0 < 128; k0 += 4) {
      v2f a = *(const v2f*)(xrow + kc * 128 + k0);       // unconditional
      const int kp = (k0 >> 1) + kh;        // K-pair within chunk
#pragma unroll
      for (int t = 0; t < 4; ++t) {
        v2f b = ((const v2f*)sW)[kp * NHID + t * 16 + l15];
        acc[t] = __builtin_amdgcn_wmma_f32_16x16x4_f32(
            false, a, false, b, (short)0, acc[t], false, false);
      }
    }
  }

  // D layout: lane L, VGPR j -> row rowBase + j + 8*(L>>4), col t*16 + (L&15)
#pragma unroll
  for (int t = 0; t < 4; ++t) {
    const int ccol = t * 16 + l15;
#pragma unroll
    for (int j = 0; j < 8; ++j) {
      int r = rowBase + j + 8 * kh;
      if (r < n) h[(size_t)r * NHID + ccol] = acc[t][j];
    }
  }
}

// -------- Stage 3: dinv[i] = rsqrt(deg); agg[i][:] = dinv^2 * h[i][:] ------
__global__ void agg_init_kernel(const float* __restrict__ h,
                                const float* __restrict__ deg,
                                float* __restrict__ dinv,
                                float* __restrict__ agg, long long total) {
  long long idx = (long long)blockIdx.x * blockDim.x + threadIdx.x;
  if (idx >= total) return;
  int i = (int)(idx >> 6);
  int j = (int)(idx & 63);
  float d = rsqrtf(deg[i]);                 // deg >= 1 always (self loop)
  if (j == 0) dinv[i] = d;
  agg[idx] = d * d * h[idx];
}

// -------- Stage 4: agg[c] += dinv[r]*dinv[c] * h[r]   (wave per edge) ------
__global__ __launch_bounds__(256)
void edge_scatter_kernel(const int* __restrict__ row,
                         const int* __restrict__ col,
                         const float* __restrict__ dinv,
                         const float* __restrict__ h,
                         float* __restrict__ agg, int e) {
  int wv   = (int)((blockIdx.x * blockDim.x + threadIdx.x) >> 5);
  int lane = threadIdx.x & 31;
  if (wv >= e) return;
  int r = __builtin_amdgcn_readfirstlane(row[wv]);
  int c = __builtin_amdgcn_readfirstlane(col[wv]);
  float norm = dinv[r] * dinv[c];
  const float* hr = h  + (size_t)r * NHID;
  float*       ac = agg + (size_t)c * NHID;
  atomicAdd(&ac[lane],      norm * hr[lane]);
  atomicAdd(&ac[lane + 32], norm * hr[lane + 32]);
}

// -------- Stage 5: per-edge [agg[c]+b1, agg[r]+b1] @ Wfc + bfc -> softmax ---
__global__ __launch_bounds__(256)
void edge_mlp_kernel(const int* __restrict__ row, const int* __restrict__ col,
                     const float* __restrict__ agg,
                     const float* __restrict__ b1,
                     const float* __restrict__ Wfc,
                     const float* __restrict__ bfc,
                     float* __restrict__ out, int e) {
  int wv   = (int)((blockIdx.x * blockDim.x + threadIdx.x) >> 5);
  int lane = threadIdx.x & 31;
  if (wv >= e) return;
  int r = __builtin_amdgcn_readfirstlane(row[wv]);
  int c = __builtin_amdgcn_readfirstlane(col[wv]);
  const float* hc = agg + (size_t)c * NHID;
  const float* hr = agg + (size_t)r * NHID;
  float a0 = 0.f, a1 = 0.f;
#pragma unroll
  for (int q = 0; q < 4; ++q) {
    int j = lane + 32 * q;                  // 0..127; branch is uniform per q
    float v = (j < 64) ? (hc[j] + b1[j]) : (hr[j - 64] + b1[j - 64]);
    a0 += v * Wfc[j * 2 + 0];
    a1 += v * Wfc[j * 2 + 1];
  }
#pragma unroll
  for (int off = 16; off > 0; off >>= 1) {  // wave32 tree reduction
    a0 += __shfl_xor(a0, off, 32);
    a1 += __shfl_xor(a1, off, 32);
  }
  if (lane == 0) {
    float l0 = a0 + bfc[0], l1 = a1 + bfc[1];
    float m  = fmaxf(l0, l1);
    float e0 = __expf(l0 - m), e1 = __expf(l1 - m);
    float s  = 1.0f / (e0 + e1);
    ((float2*)out)[wv] = make_float2(e0 * s, e1 * s);
  }
}

extern "C" void kernel_launch(void* const* d_in, const int* in_sizes, int n_in,
                              void* d_out, int out_size, void* d_ws, size_t ws_size,
                              hipStream_t stream) {
  const float* x   = (const float*)d_in[0];
  const int*   ei  = (const int*)d_in[1];
  const float* W1  = (const float*)d_in[2];
  const float* b1  = (const float*)d_in[3];
  const float* Wfc = (const float*)d_in[4];
  const float* bfc = (const float*)d_in[5];
  const int n = in_sizes[0] / NFEAT;
  const int e = in_sizes[1] / 2;
  const int* row = ei;          // edge_index[0]
  const int* col = ei + e;      // edge_index[1]

  // workspace layout: deg[n] | dinv[n] | h[n*64] | agg[n*64]  (~52 MB)
  uintptr_t p = (uintptr_t)d_ws;
  float* deg  = (float*)p;  p += (size_t)n * sizeof(float);
  float* dinv = (float*)p;  p += (size_t)n * sizeof(float);
  p = (p + 255) & ~(uintptr_t)255;
  float* h    = (float*)p;  p += (size_t)n * NHID * sizeof(float);
  p = (p + 255) & ~(uintptr_t)255;
  float* agg  = (float*)p;

  const long long aggTotal = (long long)n * NHID;

  deg_init_kernel <<<(n + 255) / 256, 256, 0, stream>>>(deg, n);
  deg_count_kernel<<<(e + 255) / 256, 256, 0, stream>>>(col, deg, e);
  gemm_x_w1_kernel<<<(n + 127) / 128, 256, 0, stream>>>(x, W1, h, n);
  agg_init_kernel <<<(unsigned)((aggTotal + 255) / 256), 256, 0, stream>>>(h, deg, dinv, agg, aggTotal);
  edge_scatter_kernel<<<(e + 7) / 8, 256, 0, stream>>>(row, col, dinv, h, agg, e);
  edge_mlp_kernel    <<<(e + 7) / 8, 256, 0, stream>>>(row, col, agg, b1, Wfc, bfc,
                                                       (float*)d_out, e);
}